// FlexibleGCN_89532888252423
// MI455X (gfx1250) — compile-verified
//
#include <hip/hip_runtime.h>

// ---------------------------------------------------------------------------
// FlexibleGCN forward for MI455X (gfx1250, wave32).
//   embedding = relu(Anorm @ (relu(Anorm @ (x@W1) + b1) @ W2) + b2)
//   logits    = embedding @ Wfc + bfc
// GEMMs use V_WMMA_F32_16X16X4_F32 (exact fp32 — GEMM FLOPs ~3.3G are
// negligible vs edge-scatter traffic, which fits in the 192MB L2 since each
// 100000x64 f32 feature matrix is 25.6MB). B panel lives in LDS interleaved
// by K-pairs so every WMMA B-fragment is a single aligned ds_load_b64.
// ---------------------------------------------------------------------------

typedef __attribute__((ext_vector_type(2))) float v2f;
typedef __attribute__((ext_vector_type(8))) float v8f;

#define HID 64

// ---- degree / normalization ------------------------------------------------

__global__ void init_deg_kernel(float* __restrict__ deg, int n) {
  int i = blockIdx.x * blockDim.x + threadIdx.x;
  if (i < n) deg[i] = 1.0f;  // self-loop contributes 1 to every node's degree
}

__global__ void deg_atomic_kernel(const long long* __restrict__ dst,
                                  float* __restrict__ deg, int e) {
  int i = blockIdx.x * blockDim.x + threadIdx.x;
  if (i < e) atomicAdd(&deg[(int)dst[i]], 1.0f);
}

__global__ void rsqrt_kernel(float* __restrict__ deg, int n) {
  int i = blockIdx.x * blockDim.x + threadIdx.x;
  if (i < n) deg[i] = rsqrtf(deg[i]);
}

// ---- fp32 WMMA GEMM: C[M,64] = A[M,K] @ B[K,64] (+ optional bias) ----------
// One wave per 16-row strip; 4 accumulators cover N=0..63; K stepped by 4.
// M must be a multiple of 16 (100000 = 6250*16).
// LDS layout: lb[(k>>1)*128 + n*2 + (k&1)]  -> fragment {B[k][n],B[k+1][n]}
// is contiguous 8 bytes, loaded as one v2f (ds_load_b64), no shuffles.

template <int K>
__global__ __launch_bounds__(256) void gemm_wmma_kernel(
    const float* __restrict__ A, const float* __restrict__ B,
    const float* __restrict__ bias, float* __restrict__ C, int M) {
  __shared__ __align__(16) float lb[K * HID];
  for (int i = threadIdx.x; i < K * HID; i += 256) {
    int k = i >> 6;       // row of B
    int n = i & 63;       // col of B
    lb[((k >> 1) << 7) + (n << 1) + (k & 1)] = B[i];
  }
  __syncthreads();

  const int lane = threadIdx.x & 31;
  const int wave = threadIdx.x >> 5;
  const int strip = blockIdx.x * 8 + wave;  // wave-uniform -> EXEC stays full
  if (strip * 16 >= M) return;

  const int half = lane >> 4;   // 0: K=k0..k0+1   1: K=k0+2..k0+3
  const int l15  = lane & 15;
  const int koff = half * 2;
  const float* arow = A + (size_t)(strip * 16 + l15) * K;

  v8f acc0 = {}, acc1 = {}, acc2 = {}, acc3 = {};
#pragma unroll 4
  for (int k0 = 0; k0 < K; k0 += 4) {
    // A fragment (16x4 f32 layout): two consecutive K values per lane.
    v2f a = *(const v2f*)(arow + k0 + koff);
    // B fragments: one aligned 8B LDS load each (K-pair interleaved layout).
    const v2f* bp = (const v2f*)&lb[(((k0 >> 1) + half)) << 7];
    acc0 = __builtin_amdgcn_wmma_f32_16x16x4_f32(false, a, false, bp[l15],
                                                 (short)0, acc0, false, false);
    acc1 = __builtin_amdgcn_wmma_f32_16x16x4_f32(false, a, false, bp[l15 + 16],
                                                 (short)0, acc1, false, false);
    acc2 = __builtin_amdgcn_wmma_f32_16x16x4_f32(false, a, false, bp[l15 + 32],
                                                 (short)0, acc2, false, false);
    acc3 = __builtin_amdgcn_wmma_f32_16x16x4_f32(false, a, false, bp[l15 + 48],
                                                 (short)0, acc3, false, false);
  }

  float bb0 = 0.f, bb1 = 0.f, bb2 = 0.f, bb3 = 0.f;
  if (bias) {
    bb0 = bias[l15];      bb1 = bias[l15 + 16];
    bb2 = bias[l15 + 32]; bb3 = bias[l15 + 48];
  }
  // C/D layout: VGPR r -> row (strip*16 + r + 8*half), col l15 (+16 per tile)
  float* crow = C + (size_t)(strip * 16 + half * 8) * HID + l15;
#pragma unroll
  for (int r = 0; r < 8; ++r) {
    float* cp = crow + (size_t)r * HID;
    cp[0]  = acc0[r] + bb0;
    cp[16] = acc1[r] + bb1;
    cp[32] = acc2[r] + bb2;
    cp[48] = acc3[r] + bb3;
  }
}

// ---- message passing (all elementwise passes are float4-vectorized) --------

// agg[i,:] = dinv[i]^2 * t[i,:]  (the appended self-loop edge; also inits agg)
__global__ void selfloop_init_kernel(const float* __restrict__ t,
                                     const float* __restrict__ dinv,
                                     float* __restrict__ agg,
                                     long long total4) {  // N*16 float4 slots
  long long i = (long long)blockIdx.x * blockDim.x + threadIdx.x;
  if (i >= total4) return;
  float d = dinv[(int)(i >> 4)];
  float s = d * d;
  const float4 v = ((const float4*)t)[i];
  ((float4*)agg)[i] = make_float4(s * v.x, s * v.y, s * v.z, s * v.w);
}

// agg[dst,:] += dinv[src]*dinv[dst] * t[src,:]
// One thread per 4 columns: float4 gather (coalesced 256B per 16 lanes) then
// 4 f32 L2 atomics to consecutive addresses. Edge indices read once per 4 cols.
__global__ void edge_scatter_kernel(const long long* __restrict__ src,
                                    const long long* __restrict__ dst,
                                    const float* __restrict__ dinv,
                                    const float* __restrict__ t,
                                    float* __restrict__ agg,
                                    long long total4) {  // E*16 float4 slots
  long long i = (long long)blockIdx.x * blockDim.x + threadIdx.x;
  if (i >= total4) return;
  int e = (int)(i >> 4);
  int c = (int)(i & 15) << 2;
  int s = (int)src[e];
  int d = (int)dst[e];
  float norm = dinv[s] * dinv[d];
  const float4 v = *(const float4*)(t + (size_t)s * HID + c);
  float* ap = agg + (size_t)d * HID + c;
  atomicAdd(ap + 0, norm * v.x);
  atomicAdd(ap + 1, norm * v.y);
  atomicAdd(ap + 2, norm * v.z);
  atomicAdd(ap + 3, norm * v.w);
}

__global__ void bias_relu_kernel(float* __restrict__ h,
                                 const float* __restrict__ b,
                                 long long total4) {  // N*16 float4 slots
  long long i = (long long)blockIdx.x * blockDim.x + threadIdx.x;
  if (i >= total4) return;
  const float4 bb = ((const float4*)b)[(int)(i & 15)];
  float4 v = ((float4*)h)[i];
  v.x = fmaxf(v.x + bb.x, 0.f);
  v.y = fmaxf(v.y + bb.y, 0.f);
  v.z = fmaxf(v.z + bb.z, 0.f);
  v.w = fmaxf(v.w + bb.w, 0.f);
  ((float4*)h)[i] = v;
}

// ---- driver ----------------------------------------------------------------

extern "C" void kernel_launch(void* const* d_in, const int* in_sizes, int n_in,
                              void* d_out, int out_size, void* d_ws,
                              size_t ws_size, hipStream_t stream) {
  const float* x        = (const float*)d_in[0];
  const float* W1       = (const float*)d_in[1];
  const float* b1       = (const float*)d_in[2];
  const float* W2       = (const float*)d_in[3];
  const float* b2       = (const float*)d_in[4];
  const float* Wfc      = (const float*)d_in[5];
  const float* bfc      = (const float*)d_in[6];
  const long long* ei   = (const long long*)d_in[7];  // int64 [2, E]

  const int N = in_sizes[0] / 128;   // 100000
  const int E = in_sizes[7] / 2;     // 1600000
  const long long* srcp = ei;
  const long long* dstp = ei + E;

  float* dinv    = (float*)d_ws;                       // N floats scratch
  float* embed   = (float*)d_out;                      // final embedding slot
  float* scratch = (float*)d_out + (size_t)N * HID;    // final logits slot

  const long long n4 = (long long)N * 16;  // float4 slots per feature matrix
  const long long e4 = (long long)E * 16;  // float4 slots over all edges
  const int gN  = (N + 255) / 256;
  const int gE  = (E + 255) / 256;
  const int gN4 = (int)((n4 + 255) / 256);
  const int gE4 = (int)((e4 + 255) / 256);
  const int gGemm = (N / 16 + 7) / 8;  // 8 waves (16-row strips) per block

  // D^{-1/2} with self-loops
  init_deg_kernel<<<gN, 256, 0, stream>>>(dinv, N);
  deg_atomic_kernel<<<gE, 256, 0, stream>>>(dstp, dinv, E);
  rsqrt_kernel<<<gN, 256, 0, stream>>>(dinv, N);

  // Layer 1: t1 = x@W1 -> scratch; aggregate into embed; +b1, relu
  gemm_wmma_kernel<128><<<gGemm, 256, 0, stream>>>(x, W1, nullptr, scratch, N);
  selfloop_init_kernel<<<gN4, 256, 0, stream>>>(scratch, dinv, embed, n4);
  edge_scatter_kernel<<<gE4, 256, 0, stream>>>(srcp, dstp, dinv, scratch, embed, e4);
  bias_relu_kernel<<<gN4, 256, 0, stream>>>(embed, b1, n4);

  // Layer 2: t2 = h1@W2 -> scratch; aggregate into embed; +b2, relu
  gemm_wmma_kernel<64><<<gGemm, 256, 0, stream>>>(embed, W2, nullptr, scratch, N);
  selfloop_init_kernel<<<gN4, 256, 0, stream>>>(scratch, dinv, embed, n4);
  edge_scatter_kernel<<<gE4, 256, 0, stream>>>(srcp, dstp, dinv, scratch, embed, e4);
  bias_relu_kernel<<<gN4, 256, 0, stream>>>(embed, b2, n4);

  // Head: logits = embedding @ Wfc + bfc -> logits slot
  gemm_wmma_kernel<64><<<gGemm, 256, 0, stream>>>(embed, Wfc, bfc, scratch, N);
}